// RecurrentSSM_65970697667397
// MI455X (gfx1250) — compile-verified
//
#include <hip/hip_runtime.h>
#include <hip/hip_bf16.h>
#include <math.h>

typedef float v2f __attribute__((ext_vector_type(2)));
typedef float v8f __attribute__((ext_vector_type(8)));

#define B_ 16
#define L_ 1024
#define D_ 1024
#define N_ 8
#define BL (B_ * L_)

__device__ __forceinline__ float soft_clamp(float x, float mn, float mx, float sharp) {
    float c = 0.5f * (mx + mn);
    float r = 0.5f * (mx - mn);
    float xn = (x - c) / (r + 1e-8f);
    return tanhf(xn * sharp) / sharp * r + c;
}

__device__ __forceinline__ float softplus_f(float v) {
    // stable softplus: max(v,0) + log1p(exp(-|v|))
    return fmaxf(v, 0.0f) + log1pf(expf(-fabsf(v)));
}

// CDNA5 hardware tanh (TRANS32). ISA requires >=1 independent op / V_NOP after a
// trans op before its output is consumed; the compiler can't see through inline
// asm so the nops are embedded.
__device__ __forceinline__ float tanh_hw(float x) {
    float r;
    asm("v_tanh_f32 %0, %1\n\tv_nop\n\tv_nop" : "=v"(r) : "v"(x));
    return r;
}

// ---------------------------------------------------------------------------
// Kernel 1: dt_pre = x(BL x D) @ W_dt^T (D x D) + b_dt, then softplus+soft_clamp.
// fp32 WMMA 16x16x4. Each wave computes a 16(M) x 64(N) strip: 4 accumulators
// reusing one A fragment per K-step.
// ---------------------------------------------------------------------------
__global__ __launch_bounds__(256) void dt_gemm_kernel(
    const float* __restrict__ x,     // (BL, D)
    const float* __restrict__ Wdt,   // (D, D) row-major (out,in)
    const float* __restrict__ bdt,   // (D)
    float* __restrict__ dt_out)      // (BL, D)
{
    const int wid_in_blk = threadIdx.x >> 5;
    const int lane       = threadIdx.x & 31;
    const int w          = blockIdx.x * 8 + wid_in_blk;   // global wave id
    // 1024 M-tiles x 16 N-groups = 16384 waves (grid sized exactly)
    const int ngroup = w & 15;        // 0..15, 64 cols each
    const int mtile  = w >> 4;        // 0..1023, 16 rows each

    const int half = lane >> 4;       // 0 or 1
    const int ln   = lane & 15;       // 0..15

    const int row  = mtile * 16 + ln;              // A-matrix M index for this lane
    const int col0 = ngroup * 64;                  // first N column of strip

    const float* __restrict__ xr = x + (size_t)row * D_;

    v8f acc0 = {}, acc1 = {}, acc2 = {}, acc3 = {};

    const float* __restrict__ w0 = Wdt + (size_t)(col0 +  0 + ln) * D_;
    const float* __restrict__ w1 = Wdt + (size_t)(col0 + 16 + ln) * D_;
    const float* __restrict__ w2 = Wdt + (size_t)(col0 + 32 + ln) * D_;
    const float* __restrict__ w3 = Wdt + (size_t)(col0 + 48 + ln) * D_;

    #pragma unroll 4
    for (int k0 = 0; k0 < D_; k0 += 4) {
        const int kk = k0 + 2 * half;              // this lane's K pair
        v2f a  = *(const v2f*)(xr + kk);           // A[m][kk], A[m][kk+1]
        v2f b0 = *(const v2f*)(w0 + kk);           // B[kk][n] = Wdt[n][kk]
        v2f b1 = *(const v2f*)(w1 + kk);
        v2f b2 = *(const v2f*)(w2 + kk);
        v2f b3 = *(const v2f*)(w3 + kk);
        acc0 = __builtin_amdgcn_wmma_f32_16x16x4_f32(false, a, false, b0, (short)0, acc0, false, false);
        acc1 = __builtin_amdgcn_wmma_f32_16x16x4_f32(false, a, false, b1, (short)0, acc1, false, false);
        acc2 = __builtin_amdgcn_wmma_f32_16x16x4_f32(false, a, false, b2, (short)0, acc2, false, false);
        acc3 = __builtin_amdgcn_wmma_f32_16x16x4_f32(false, a, false, b3, (short)0, acc3, false, false);
    }

    // Epilogue: d-reg v holds row M = mtile*16 + v + 8*half, col = col0 + j*16 + ln
    #pragma unroll
    for (int j = 0; j < 4; ++j) {
        v8f acc = (j == 0) ? acc0 : (j == 1) ? acc1 : (j == 2) ? acc2 : acc3;
        const int col = col0 + j * 16 + ln;
        const float bias = bdt[col];
        #pragma unroll
        for (int v = 0; v < 8; ++v) {
            const int r = mtile * 16 + v + 8 * half;
            float val = acc[v] + bias;
            float sp  = softplus_f(val);
            float dt  = soft_clamp(sp, 1e-3f, 0.1f, 0.5f);
            dt_out[(size_t)r * D_ + col] = dt;
        }
    }
}

// ---------------------------------------------------------------------------
// Kernel 2: Bp = x @ W_B^T, Cp = x @ W_C^T  -> (BL, 8) each.
// One wave per row; lanes stride K; cross-lane reduce with shfl_xor.
// ---------------------------------------------------------------------------
__global__ __launch_bounds__(256) void proj_bc_kernel(
    const float* __restrict__ x,   // (BL, D)
    const float* __restrict__ WB,  // (N, D)
    const float* __restrict__ WC,  // (N, D)
    float* __restrict__ Bp,        // (BL, N)
    float* __restrict__ Cp)        // (BL, N)
{
    const int row  = (blockIdx.x * blockDim.x + threadIdx.x) >> 5;
    const int lane = threadIdx.x & 31;
    if (row >= BL) return;

    const float* __restrict__ xr = x + (size_t)row * D_;
    float accB[N_], accC[N_];
    #pragma unroll
    for (int n = 0; n < N_; ++n) { accB[n] = 0.0f; accC[n] = 0.0f; }

    for (int k = lane; k < D_; k += 32) {
        float xv = xr[k];
        #pragma unroll
        for (int n = 0; n < N_; ++n) {
            accB[n] = fmaf(xv, WB[n * D_ + k], accB[n]);
            accC[n] = fmaf(xv, WC[n * D_ + k], accC[n]);
        }
    }
    #pragma unroll
    for (int n = 0; n < N_; ++n) {
        for (int off = 16; off > 0; off >>= 1) {
            accB[n] += __shfl_xor(accB[n], off, 32);
            accC[n] += __shfl_xor(accC[n], off, 32);
        }
    }
    if (lane == 0) {
        #pragma unroll
        for (int n = 0; n < N_; ++n) {
            Bp[(size_t)row * N_ + n] = accB[n];
            Cp[(size_t)row * N_ + n] = accC[n];
        }
    }
}

// ---------------------------------------------------------------------------
// Kernel 3: A = -exp(soft_clamp(A_log, -5, 5, 0.5))   (D x N)
// ---------------------------------------------------------------------------
__global__ void a_xform_kernel(const float* __restrict__ A_log, float* __restrict__ An) {
    int i = blockIdx.x * blockDim.x + threadIdx.x;
    if (i < D_ * N_) {
        An[i] = -expf(soft_clamp(A_log[i], -5.0f, 5.0f, 0.5f));
    }
}

// ---------------------------------------------------------------------------
// Kernel 4: sequential scan. One workgroup (1024 threads) per batch.
// Thread d owns h[d][0..7] in registers. Renorm every 10 steps via LDS tree.
// Critical-path transcendentals use HW v_tanh_f32 / v_exp_f32.
// ---------------------------------------------------------------------------
__global__ __launch_bounds__(1024) void scan_kernel(
    const float* __restrict__ x,    // (B, L, D)
    const float* __restrict__ dt,   // (B, L, D)
    const float* __restrict__ Bp,   // (B, L, N)
    const float* __restrict__ Cp,   // (B, L, N)
    const float* __restrict__ An,   // (D, N)  (already -exp(clamp(...)))
    const float* __restrict__ Dp,   // (D)
    float* __restrict__ out)        // (B, L, D)
{
    const int b = blockIdx.x;
    const int d = threadIdx.x;
    const int wid  = threadIdx.x >> 5;
    const int lane = threadIdx.x & 31;

    __shared__ float red[32];
    __shared__ float s_scale;

    float A_[N_];
    #pragma unroll
    for (int n = 0; n < N_; ++n) A_[n] = An[d * N_ + n];
    const float Dpar = Dp[d];

    float h[N_];
    #pragma unroll
    for (int n = 0; n < N_; ++n) h[n] = 0.0f;

    const float* __restrict__ xb  = x  + (size_t)b * L_ * D_;
    const float* __restrict__ dtb = dt + (size_t)b * L_ * D_;
    const float* __restrict__ bpb = Bp + (size_t)b * L_ * N_;
    const float* __restrict__ cpb = Cp + (size_t)b * L_ * N_;
    float* __restrict__ ob        = out + (size_t)b * L_ * D_;

    // soft_clamp(v,-8,8,0.5) == 16 * tanh( v * 0.5/(8+1e-8) )
    const float kArg = 0.5f / (8.0f + 1e-8f);

    int renorm_ctr = 0;   // renorm when 0 (t = 0, 10, 20, ...)

    for (int t = 0; t < L_; ++t) {
        const float xv  = xb [(size_t)t * D_ + d];
        const float dtv = dtb[(size_t)t * D_ + d];

        if (t + 1 < L_) {   // CDNA5 global_prefetch for next timestep
            __builtin_prefetch(xb  + (size_t)(t + 1) * D_ + d, 0, 0);
            __builtin_prefetch(dtb + (size_t)(t + 1) * D_ + d, 0, 0);
        }

        float Bt[N_], Ct[N_];
        #pragma unroll
        for (int n = 0; n < N_; ++n) {
            Bt[n] = bpb[(size_t)t * N_ + n];
            Ct[n] = cpb[(size_t)t * N_ + n];
        }

        const float dx = dtv * xv;
        #pragma unroll
        for (int n = 0; n < N_; ++n) {
            float v  = dtv * A_[n];
            float sc = 16.0f * tanh_hw(v * kArg);   // soft_clamp(v,-8,8,0.5)
            float dA = __expf(sc);                  // v_exp_f32 path
            h[n] = fmaf(h[n], dA, dx * Bt[n]);
        }

        if (renorm_ctr == 0) {
            renorm_ctr = 10;
            // sum of squares over all (d,n) of this batch
            float ss = 0.0f;
            #pragma unroll
            for (int n = 0; n < N_; ++n) ss = fmaf(h[n], h[n], ss);
            for (int off = 16; off > 0; off >>= 1) ss += __shfl_xor(ss, off, 32);
            if (lane == 0) red[wid] = ss;
            __syncthreads();
            if (threadIdx.x == 0) {
                float tot = 0.0f;
                #pragma unroll
                for (int i = 0; i < 32; ++i) tot += red[i];
                float hn = sqrtf(tot);
                s_scale = fminf(20.0f / (hn + 1e-8f), 1.0f);
            }
            __syncthreads();
            const float sc = s_scale;
            #pragma unroll
            for (int n = 0; n < N_; ++n) h[n] *= sc;
            // NOTE: no third barrier needed — the next renorm's first
            // __syncthreads() orders these s_scale reads against its rewrite.
        }
        renorm_ctr--;

        float y = Dpar * xv;
        #pragma unroll
        for (int n = 0; n < N_; ++n) y = fmaf(h[n], Ct[n], y);
        ob[(size_t)t * D_ + d] = y;
    }
}

// ---------------------------------------------------------------------------
extern "C" void kernel_launch(void* const* d_in, const int* in_sizes, int n_in,
                              void* d_out, int out_size, void* d_ws, size_t ws_size,
                              hipStream_t stream) {
    const float* x     = (const float*)d_in[0];  // (B,L,D)
    const float* W_dt  = (const float*)d_in[1];  // (D,D)
    const float* b_dt  = (const float*)d_in[2];  // (D)
    const float* W_B   = (const float*)d_in[3];  // (N,D)
    const float* W_C   = (const float*)d_in[4];  // (N,D)
    const float* A_log = (const float*)d_in[5];  // (D,N)
    const float* D_par = (const float*)d_in[6];  // (D)
    float* out = (float*)d_out;

    char* ws = (char*)d_ws;
    float* dt = (float*)ws;                                        // BL*D floats = 64 MB
    float* Bp = (float*)(ws + (size_t)BL * D_ * 4);                // BL*N floats
    float* Cp = (float*)(ws + (size_t)BL * D_ * 4 + (size_t)BL * N_ * 4);
    float* An = (float*)(ws + (size_t)BL * D_ * 4 + 2 * (size_t)BL * N_ * 4);

    // 1) dt = soft_clamp(softplus(x @ W_dt^T + b_dt)) via fp32 WMMA
    //    16384 waves = 1024 M-tiles x 16 N-groups; 8 waves/block -> 2048 blocks
    dt_gemm_kernel<<<2048, 256, 0, stream>>>(x, W_dt, b_dt, dt);

    // 2) Bp / Cp projections: one wave per row -> 16384 waves -> 2048 blocks
    proj_bc_kernel<<<2048, 256, 0, stream>>>(x, W_B, W_C, Bp, Cp);

    // 3) A transform (8192 elements)
    a_xform_kernel<<<(D_ * N_ + 255) / 256, 256, 0, stream>>>(A_log, An);

    // 4) sequential scan: one workgroup per batch
    scan_kernel<<<B_, 1024, 0, stream>>>(x, dt, Bp, Cp, An, D_par, out);
}